// GraphEncoder_3178275799160
// MI455X (gfx1250) — compile-verified
//
#include <hip/hip_runtime.h>
#include <hip/hip_bf16.h>
#include <math.h>

// Model constants (match reference)
#define GH    4      // heads
#define GC    128    // per-head dim
#define GHID  128    // hidden
#define GHC   512    // H*C
#define GL    3      // layers
#define GNF   5      // node feats
#define GB    32     // graphs

typedef float v2f __attribute__((ext_vector_type(2)));
typedef float v8f __attribute__((ext_vector_type(8)));

// ---------------------------------------------------------------------------
// Utility: grid-stride fill
// ---------------------------------------------------------------------------
__global__ void fill_f32(float* __restrict__ p, float v, long long n) {
    long long i = (long long)blockIdx.x * blockDim.x + threadIdx.x;
    long long stride = (long long)gridDim.x * blockDim.x;
    for (; i < n; i += stride) p[i] = v;
}

// ---------------------------------------------------------------------------
// Input projection: h = relu(x @ node_W + node_b)   [N,NF]@[NF,HID]
// ---------------------------------------------------------------------------
__global__ void input_proj(const float* __restrict__ x,
                           const float* __restrict__ W,
                           const float* __restrict__ b,
                           float* __restrict__ h, int N) {
    int t = blockIdx.x * blockDim.x + threadIdx.x;
    if (t >= N * GHID) return;
    int n = t >> 7, j = t & 127;
    float acc = b[j];
    #pragma unroll
    for (int f = 0; f < GNF; ++f)
        acc += x[n * GNF + f] * W[f * GHID + j];
    h[t] = fmaxf(acc, 0.0f);
}

// ---------------------------------------------------------------------------
// WMMA fp32 GEMM: Out[N,512] = A[N,128] @ W[128,512]
// One wave32 per 16x32 output strip (two 16x16 tiles sharing the A fragment).
// K accumulated with V_WMMA_F32_16X16X4_F32 (exact fp32 math on matrix pipe).
// A-matrix VGPR layout (ISA 7.12.2, 32-bit A 16x4): lanes 0-15 hold K=k,k+1,
// lanes 16-31 hold K=k+2,k+3 (one per VGPR); M = lane&15.  B mirrors with N
// across lanes.  C/D: VGPR r -> (M = r + 8*laneHalf, N = lane&15).
//
// Tail handling: A row m only ever contributes to D row m, so for a partial
// last M-tile we just CLAMP the load row (stay in bounds); stores take a
// wave-uniform branch: full tiles get straight-line stores, only the last
// partial tile pays per-row guards.
// ---------------------------------------------------------------------------
__global__ __launch_bounds__(256)
void gemm_wmma_f32(const float* __restrict__ A,
                   const float* __restrict__ W,
                   float* __restrict__ Out, int N) {
    const int K = GHID, NC = GHC;
    int wave = (blockIdx.x * blockDim.x + threadIdx.x) >> 5;
    int lane = threadIdx.x & 31;
    const int nstrips = NC / 32;                // 16 strips of 32 cols
    int m0 = (wave / nstrips) * 16;
    int n0 = (wave % nstrips) * 32;
    if (m0 >= N) return;                        // wave-uniform exit

    int half = lane >> 4;                       // 0: lanes 0-15, 1: lanes 16-31
    int l16  = lane & 15;
    int arow = m0 + l16;
    if (arow >= N) arow = N - 1;                // clamp: row never stored anyway

    const float2* __restrict__ Arow = (const float2*)(A + (size_t)arow * K);

    v8f acc0 = {}, acc1 = {};
    #pragma unroll 4
    for (int k = 0; k < K; k += 4) {
        int ka = k + half * 2;                  // even -> one b64 load
        float2 apair = Arow[ka >> 1];
        v2f a; a.x = apair.x; a.y = apair.y;

        const float* __restrict__ w0 = W + (size_t)ka * NC + n0 + l16;
        v2f b0, b1;
        b0.x = w0[0];        b0.y = w0[NC];         // cols n0..n0+15
        b1.x = w0[16];       b1.y = w0[NC + 16];    // cols n0+16..n0+31

        acc0 = __builtin_amdgcn_wmma_f32_16x16x4_f32(
                 false, a, false, b0, (short)0, acc0, false, false);
        acc1 = __builtin_amdgcn_wmma_f32_16x16x4_f32(
                 false, a, false, b1, (short)0, acc1, false, false);
    }

    // Store: VGPR r holds row (m0 + r + 8*half), col n0 + l16 (+16 for acc1).
    float* __restrict__ base = Out + (size_t)(m0 + half * 8) * NC + n0 + l16;
    if (m0 + 16 <= N) {                         // wave-uniform: full tile
        #pragma unroll
        for (int r = 0; r < 8; ++r) {
            base[(size_t)r * NC]      = acc0[r];
            base[(size_t)r * NC + 16] = acc1[r];
        }
    } else {                                    // partial last tile
        #pragma unroll
        for (int r = 0; r < 8; ++r) {
            if (m0 + r + half * 8 < N) {
                base[(size_t)r * NC]      = acc0[r];
                base[(size_t)r * NC + 16] = acc1[r];
            }
        }
    }
}

// ---------------------------------------------------------------------------
// Attention scores: a_src[n,h] = <hp[n,h,:], att_src[h,:]>, same for dst
// ---------------------------------------------------------------------------
__global__ void att_scores(const float* __restrict__ hp,
                           const float* __restrict__ att_src,
                           const float* __restrict__ att_dst,
                           float* __restrict__ a_src,
                           float* __restrict__ a_dst, int N) {
    int t = blockIdx.x * blockDim.x + threadIdx.x;
    if (t >= N * GH) return;
    int n = t >> 2, hh = t & 3;
    const float* row = hp + (size_t)n * GHC + hh * GC;
    const float* as  = att_src + hh * GC;
    const float* ad  = att_dst + hh * GC;
    float ss = 0.f, sd = 0.f;
    #pragma unroll 4
    for (int c = 0; c < GC; ++c) { float v = row[c]; ss += v * as[c]; sd += v * ad[c]; }
    a_src[t] = ss;
    a_dst[t] = sd;
}

// Float atomic max via monotone int-bits trick (init buffer to -inf)
__device__ inline void atomicMaxF(float* addr, float v) {
    if (v >= 0.0f) atomicMax((int*)addr, __float_as_int(v));
    else           atomicMin((unsigned int*)addr, __float_as_uint(v));
}

// ---------------------------------------------------------------------------
// Edge pass 1: e = leaky_relu(a_src[s]+a_dst[d], 0.2); segment-max into m[dst]
// Edges E..E+N-1 are the PyG self-loops (s = d = eid - E).
// ---------------------------------------------------------------------------
__global__ void edge_logits(const float* __restrict__ a_src,
                            const float* __restrict__ a_dst,
                            const int* __restrict__ src,
                            const int* __restrict__ dst,
                            int E, int N,
                            float* __restrict__ ebuf,
                            float* __restrict__ m) {
    int t = blockIdx.x * blockDim.x + threadIdx.x;
    int total = (E + N) * GH;
    if (t >= total) return;
    int eid = t >> 2, hh = t & 3;
    int s, d;
    if (eid < E) { s = src[eid]; d = dst[eid]; } else { s = d = eid - E; }
    float e = a_src[s * GH + hh] + a_dst[d * GH + hh];
    e = (e > 0.0f) ? e : 0.2f * e;
    ebuf[t] = e;
    atomicMaxF(&m[d * GH + hh], e);
}

// ---------------------------------------------------------------------------
// Edge pass 2: ex = exp(e - m[dst]); segment-sum into denom[dst]
// ---------------------------------------------------------------------------
__global__ void edge_exp(const int* __restrict__ src,
                         const int* __restrict__ dst,
                         int E, int N,
                         float* __restrict__ ebuf,
                         const float* __restrict__ m,
                         float* __restrict__ denom) {
    int t = blockIdx.x * blockDim.x + threadIdx.x;
    int total = (E + N) * GH;
    if (t >= total) return;
    int eid = t >> 2, hh = t & 3;
    int d = (eid < E) ? dst[eid] : (eid - E);
    float ex = expf(ebuf[t] - m[d * GH + hh]);
    ebuf[t] = ex;
    atomicAdd(&denom[d * GH + hh], ex);
}

// ---------------------------------------------------------------------------
// Edge pass 3: accum[dst,h,c] += hp[src,h,c] * alpha   (one thread per elem)
// ---------------------------------------------------------------------------
__global__ void edge_scatter(const float* __restrict__ hp,
                             const float* __restrict__ ebuf,
                             const float* __restrict__ denom,
                             const int* __restrict__ src,
                             const int* __restrict__ dst,
                             int E, int N,
                             float* __restrict__ accum) {
    long long t = (long long)blockIdx.x * blockDim.x + threadIdx.x;
    long long total = (long long)(E + N) * GHC;
    if (t >= total) return;
    int eid = (int)(t >> 9);
    int j   = (int)(t & (GHC - 1));
    int s, d;
    if (eid < E) { s = src[eid]; d = dst[eid]; } else { s = d = eid - E; }
    int hh = j >> 7;
    float alpha = ebuf[eid * GH + hh] / (denom[d * GH + hh] + 1e-16f);
    atomicAdd(&accum[(size_t)d * GHC + j], hp[(size_t)s * GHC + j] * alpha);
}

// ---------------------------------------------------------------------------
// Per-node epilogue: head-mean + bias -> LayerNorm -> relu + residual
// One 128-thread block per node; LDS tree reduction for mean/var.
// ---------------------------------------------------------------------------
__global__ __launch_bounds__(GHID)
void node_finish(const float* __restrict__ accum,
                 const float* __restrict__ bias,
                 const float* __restrict__ lng,
                 const float* __restrict__ lnb,
                 float* __restrict__ h, int N) {
    int n = blockIdx.x;
    int j = threadIdx.x;
    if (n >= N) return;
    const float* ac = accum + (size_t)n * GHC;
    float v = 0.25f * (ac[j] + ac[GC + j] + ac[2 * GC + j] + ac[3 * GC + j]) + bias[j];

    __shared__ float red[GHID];
    red[j] = v; __syncthreads();
    #pragma unroll
    for (int s = GHID / 2; s > 0; s >>= 1) { if (j < s) red[j] += red[j + s]; __syncthreads(); }
    float mu = red[0] * (1.0f / GHID);
    __syncthreads();
    float dvv = v - mu;
    red[j] = dvv * dvv; __syncthreads();
    #pragma unroll
    for (int s = GHID / 2; s > 0; s >>= 1) { if (j < s) red[j] += red[j + s]; __syncthreads(); }
    float var = red[0] * (1.0f / GHID);

    float y = dvv * rsqrtf(var + 1e-5f) * lng[j] + lnb[j];
    size_t idx = (size_t)n * GHID + j;
    h[idx] = fmaxf(y, 0.0f) + h[idx];
}

// ---------------------------------------------------------------------------
// Global mean pool
// ---------------------------------------------------------------------------
__global__ void pool_sum(const float* __restrict__ h,
                         const int* __restrict__ batch,
                         float* __restrict__ sums,
                         float* __restrict__ cnts, int N) {
    int t = blockIdx.x * blockDim.x + threadIdx.x;
    if (t >= N * GHID) return;
    int n = t >> 7, j = t & 127;
    int b = batch[n];
    atomicAdd(&sums[b * GHID + j], h[t]);
    if (j == 0) atomicAdd(&cnts[b], 1.0f);
}

__global__ void pool_div(const float* __restrict__ sums,
                         const float* __restrict__ cnts,
                         float* __restrict__ emb) {
    int t = blockIdx.x * blockDim.x + threadIdx.x;
    if (t >= GB * GHID) return;
    int b = t >> 7;
    emb[t] = sums[t] / fmaxf(cnts[b], 1.0f);
}

// ---------------------------------------------------------------------------
// Host launcher
// ---------------------------------------------------------------------------
extern "C" void kernel_launch(void* const* d_in, const int* in_sizes, int n_in,
                              void* d_out, int out_size, void* d_ws, size_t ws_size,
                              hipStream_t stream) {
    const float* x        = (const float*)d_in[0];
    const int*   edge_idx = (const int*)  d_in[1];
    const int*   batch    = (const int*)  d_in[2];
    const float* node_W   = (const float*)d_in[3];
    const float* node_b   = (const float*)d_in[4];
    const float* Ws       = (const float*)d_in[5];
    const float* att_srcs = (const float*)d_in[6];
    const float* att_dsts = (const float*)d_in[7];
    const float* biases   = (const float*)d_in[8];
    const float* ln_gs    = (const float*)d_in[9];
    const float* ln_bs    = (const float*)d_in[10];

    const int N = in_sizes[0] / GNF;
    const int E = in_sizes[1] / 2;
    const int Etot = E + N;
    const int* src = edge_idx;
    const int* dst = edge_idx + E;

    // Outputs: h [N,128] then graph_emb [32,128]
    float* h   = (float*)d_out;
    float* emb = h + (size_t)N * GHID;

    // Workspace layout
    float* ws     = (float*)d_ws;
    float* hp     = ws; ws += (size_t)N * GHC;     // projected features [N,H,C]
    float* accum  = ws; ws += (size_t)N * GHC;     // message sums       [N,H,C]
    float* a_src  = ws; ws += (size_t)N * GH;
    float* a_dst  = ws; ws += (size_t)N * GH;
    float* mbuf   = ws; ws += (size_t)N * GH;
    float* denom  = ws; ws += (size_t)N * GH;
    float* ebuf   = ws; ws += (size_t)Etot * GH;
    float* sums   = ws; ws += GB * GHID;
    float* cnts   = ws; ws += GB;

    const int BLK = 256;
    const float NEG_INF = -__builtin_inff();

    // 1. input projection
    input_proj<<<(N * GHID + BLK - 1) / BLK, BLK, 0, stream>>>(x, node_W, node_b, h, N);

    // GEMM grid: one wave per 16x32 output strip
    int mtiles = (N + 15) / 16;
    int waves  = mtiles * (GHC / 32);
    int gblk   = (waves * 32 + BLK - 1) / BLK;

    for (int l = 0; l < GL; ++l) {
        const float* W   = Ws       + (size_t)l * GHID * GHC;
        const float* as_ = att_srcs + (size_t)l * GH * GC;
        const float* ad_ = att_dsts + (size_t)l * GH * GC;
        const float* bi  = biases   + (size_t)l * GHID;
        const float* lg  = ln_gs    + (size_t)l * GHID;
        const float* lb  = ln_bs    + (size_t)l * GHID;

        // init segment buffers
        fill_f32<<<1024, BLK, 0, stream>>>(mbuf, NEG_INF, (long long)N * GH);
        fill_f32<<<1024, BLK, 0, stream>>>(denom, 0.0f,   (long long)N * GH);
        fill_f32<<<2048, BLK, 0, stream>>>(accum, 0.0f,   (long long)N * GHC);

        // dense projection via fp32 WMMA
        gemm_wmma_f32<<<gblk, BLK, 0, stream>>>(h, W, hp, N);

        // attention coefficients
        att_scores<<<(N * GH + BLK - 1) / BLK, BLK, 0, stream>>>(hp, as_, ad_, a_src, a_dst, N);

        // edge softmax (max / exp / sum), then weighted scatter
        int ethreads = Etot * GH;
        edge_logits<<<(ethreads + BLK - 1) / BLK, BLK, 0, stream>>>(a_src, a_dst, src, dst, E, N, ebuf, mbuf);
        edge_exp   <<<(ethreads + BLK - 1) / BLK, BLK, 0, stream>>>(src, dst, E, N, ebuf, mbuf, denom);
        long long sthreads = (long long)Etot * GHC;
        edge_scatter<<<(unsigned)((sthreads + BLK - 1) / BLK), BLK, 0, stream>>>(
            hp, ebuf, denom, src, dst, E, N, accum);

        // head-mean + bias + LN + relu + residual
        node_finish<<<N, GHID, 0, stream>>>(accum, bi, lg, lb, h, N);
    }

    // 2. global mean pool
    fill_f32<<<16, BLK, 0, stream>>>(sums, 0.0f, (long long)GB * GHID);
    fill_f32<<<1, 32, 0, stream>>>(cnts, 0.0f, (long long)GB);
    pool_sum<<<(N * GHID + BLK - 1) / BLK, BLK, 0, stream>>>(h, batch, sums, cnts, N);
    pool_div<<<(GB * GHID + BLK - 1) / BLK, BLK, 0, stream>>>(sums, cnts, emb);
}